// NadarayaWatson_34153579937939
// MI455X (gfx1250) — compile-verified
//
#include <hip/hip_runtime.h>
#include <math.h>

typedef __attribute__((ext_vector_type(2))) float v2f;
typedef __attribute__((ext_vector_type(8))) float v8f;

#define D_DIM 512
#define B_DIM 128
#define N_DIM 4096
#define INV_2BW2 3.125f  // 1 / (2 * 0.4^2)

// workspace layout (float offsets)
#define WS_DIFF   0
#define WS_X1N    (N_DIM * D_DIM)
#define WS_X0N    (WS_X1N + N_DIM)
#define WS_X10    (WS_X0N + N_DIM)
#define WS_LOGITS (WS_X10 + N_DIM)

static __device__ __forceinline__ v8f wmma_f32(v2f a, v2f b, v8f c) {
    // D = A(16x4) * B(4x16) + C(16x16), full fp32
    return __builtin_amdgcn_wmma_f32_16x16x4_f32(
        /*neg_a=*/false, a, /*neg_b=*/false, b,
        /*c_mod=*/(short)0, c, /*reuse_a=*/false, /*reuse_b=*/false);
}

// ---------------------------------------------------------------------------
// Phase 1: gather x0/x1 rows, emit diff = x1 - x0, and row stats x1n,x0n,x10
// grid = N_DIM blocks of 128 threads
// ---------------------------------------------------------------------------
__global__ void nw_prep_kernel(const float* __restrict__ pi0,
                               const float* __restrict__ pi1,
                               const int* __restrict__ idx,
                               float* __restrict__ diff,
                               float* __restrict__ x1n,
                               float* __restrict__ x0n,
                               float* __restrict__ x10) {
    const int n   = blockIdx.x;
    const int tid = threadIdx.x;
    const size_t row = (size_t)idx[n] * D_DIM;
    const float* __restrict__ r0 = pi0 + row;
    const float* __restrict__ r1 = pi1 + row;
    float* __restrict__ dn = diff + (size_t)n * D_DIM;

    float s1 = 0.f, s0 = 0.f, s10 = 0.f;
    for (int d = tid; d < D_DIM; d += 128) {
        float v1 = r1[d];
        float v0 = r0[d];
        dn[d] = v1 - v0;
        s1  += v1 * v1;
        s0  += v0 * v0;
        s10 += v1 * v0;
    }
    __shared__ float a1[128], a0[128], a10[128];
    a1[tid] = s1; a0[tid] = s0; a10[tid] = s10;
    __syncthreads();
    for (int off = 64; off > 0; off >>= 1) {
        if (tid < off) {
            a1[tid]  += a1[tid + off];
            a0[tid]  += a0[tid + off];
            a10[tid] += a10[tid + off];
        }
        __syncthreads();
    }
    if (tid == 0) {
        x1n[n] = a1[0];
        x0n[n] = a0[0];
        x10[n] = a10[0];
    }
}

// ---------------------------------------------------------------------------
// Phase 2: logits[b][n] = (2*(a*zx1 + bb*zx0) - (a^2 x1n + 2ab x10 + b^2 x0n)) / 0.32
// (the z2[b] row-constant is dropped: softmax is shift invariant per row)
// block = 128 threads (4 waves), each wave owns one 16-wide N tile
// grid  = (64 n-groups of 64, 8 m-tiles of 16)
// ---------------------------------------------------------------------------
__global__ void nw_logits_kernel(const float* __restrict__ z_t,
                                 const float* __restrict__ tvec,
                                 const float* __restrict__ pi0,
                                 const float* __restrict__ pi1,
                                 const int* __restrict__ idx,
                                 const float* __restrict__ x1n,
                                 const float* __restrict__ x0n,
                                 const float* __restrict__ x10,
                                 float* __restrict__ logits) {
    __shared__ float z_lds[16 * D_DIM];   // 32 KB: 16 rows of z_t, row-major
    __shared__ float t_lds[16];

    const int tid  = threadIdx.x;
    const int wave = tid >> 5;
    const int lane = tid & 31;
    const int m0   = blockIdx.y * 16;

    // stage z tile (rows m0..m0+15 are contiguous in memory)
    const float* zsrc = z_t + (size_t)m0 * D_DIM;
    for (int i = tid; i < 16 * D_DIM; i += 128) z_lds[i] = zsrc[i];
    if (tid < 16) t_lds[tid] = tvec[m0 + tid];
    __syncthreads();

    const int n0   = blockIdx.x * 64 + wave * 16;
    const int nl   = lane & 15;            // column within tile / A row
    const int koff = (lane >> 4) * 2;      // K sub-offset per A/B layout
    const int n    = n0 + nl;

    const size_t rowbase = (size_t)idx[n] * D_DIM;
    const float* __restrict__ b1p = pi1 + rowbase;
    const float* __restrict__ b0p = pi0 + rowbase;
    const float* __restrict__ ap  = &z_lds[nl * D_DIM];

    v8f acc1 = {}; v8f acc0 = {};
    for (int k0 = 0; k0 < D_DIM; k0 += 4) {
        const int k = k0 + koff;
        v2f av = *(const v2f*)(ap + k);           // ds_load_b64
        v2f b1 = *(const v2f*)(b1p + k);          // global_load_b64
        v2f b0 = *(const v2f*)(b0p + k);          // global_load_b64
        acc1 = wmma_f32(av, b1, acc1);
        acc0 = wmma_f32(av, b0, acc0);
    }

    const float w1 = x1n[n], w0 = x0n[n], w10 = x10[n];
    const int mh = (lane >> 4) * 8;
    #pragma unroll
    for (int r = 0; r < 8; ++r) {
        const int ml = r + mh;
        const float aB = t_lds[ml];
        const float bB = 1.0f - aB;
        const float c  = aB * aB * w1 + 2.0f * aB * bB * w10 + bB * bB * w0;
        const float lg = (2.0f * (aB * acc1[r] + bB * acc0[r]) - c) * INV_2BW2;
        logits[(size_t)(m0 + ml) * N_DIM + n] = lg;
    }
}

// ---------------------------------------------------------------------------
// Phase 3: in-place row softmax over N_DIM. grid = B_DIM, block = 256
// ---------------------------------------------------------------------------
__global__ void nw_softmax_kernel(float* __restrict__ logits) {
    __shared__ float red[256];
    const int tid = threadIdx.x;
    float* __restrict__ L = logits + (size_t)blockIdx.x * N_DIM;

    float mx = -INFINITY;
    for (int i = tid; i < N_DIM; i += 256) mx = fmaxf(mx, L[i]);
    red[tid] = mx;
    __syncthreads();
    for (int off = 128; off > 0; off >>= 1) {
        if (tid < off) red[tid] = fmaxf(red[tid], red[tid + off]);
        __syncthreads();
    }
    const float rowmax = red[0];
    __syncthreads();

    float s = 0.f;
    for (int i = tid; i < N_DIM; i += 256) s += __expf(L[i] - rowmax);
    red[tid] = s;
    __syncthreads();
    for (int off = 128; off > 0; off >>= 1) {
        if (tid < off) red[tid] += red[tid + off];
        __syncthreads();
    }
    const float inv = 1.0f / red[0];
    __syncthreads();

    for (int i = tid; i < N_DIM; i += 256) L[i] = __expf(L[i] - rowmax) * inv;
}

// ---------------------------------------------------------------------------
// Phase 4: out = P(128x4096) @ diff(4096x512) via f32 WMMA, K = 4096
// block = 128 threads (4 waves), each wave owns one 16-wide D tile
// grid  = (8 d-groups of 4 tiles, 8 m-tiles)
// ---------------------------------------------------------------------------
__global__ void nw_out_kernel(const float* __restrict__ probs,
                              const float* __restrict__ diff,
                              float* __restrict__ out) {
    const int tid  = threadIdx.x;
    const int wave = tid >> 5;
    const int lane = tid & 31;
    const int m0   = blockIdx.y * 16;
    const int d0   = (blockIdx.x * 4 + wave) * 16;
    const int dl   = lane & 15;
    const int koff = (lane >> 4) * 2;
    const int d    = d0 + dl;

    const float* __restrict__ ap = probs + (size_t)(m0 + dl) * N_DIM; // A row m = lane&15
    const float* __restrict__ bp = diff + d;

    v8f acc = {};
    for (int k0 = 0; k0 < N_DIM; k0 += 4) {
        const int k = k0 + koff;
        v2f av = *(const v2f*)(ap + k);                    // P[m][k..k+1]
        v2f bv;
        bv.x = bp[(size_t)k * D_DIM];                      // diff[k][d]
        bv.y = bp[(size_t)(k + 1) * D_DIM];                // diff[k+1][d]
        acc = wmma_f32(av, bv, acc);
    }

    const int mh = (lane >> 4) * 8;
    #pragma unroll
    for (int r = 0; r < 8; ++r) {
        out[(size_t)(m0 + r + mh) * D_DIM + d] = acc[r];
    }
}

// ---------------------------------------------------------------------------
extern "C" void kernel_launch(void* const* d_in, const int* in_sizes, int n_in,
                              void* d_out, int out_size, void* d_ws, size_t ws_size,
                              hipStream_t stream) {
    const float* z_t = (const float*)d_in[0];   // (128, 512)
    const float* t   = (const float*)d_in[1];   // (128,)
    const float* pi0 = (const float*)d_in[2];   // (50000, 512)
    const float* pi1 = (const float*)d_in[3];   // (50000, 512)
    const int*   idx = (const int*)d_in[4];     // (4096,)
    float* out = (float*)d_out;                 // (128, 512)

    float* ws     = (float*)d_ws;
    float* diff   = ws + WS_DIFF;
    float* x1n    = ws + WS_X1N;
    float* x0n    = ws + WS_X0N;
    float* x10    = ws + WS_X10;
    float* logits = ws + WS_LOGITS;

    nw_prep_kernel<<<N_DIM, 128, 0, stream>>>(pi0, pi1, idx, diff, x1n, x0n, x10);
    nw_logits_kernel<<<dim3(64, 8), 128, 0, stream>>>(z_t, t, pi0, pi1, idx,
                                                      x1n, x0n, x10, logits);
    nw_softmax_kernel<<<B_DIM, 256, 0, stream>>>(logits);
    nw_out_kernel<<<dim3(8, 8), 128, 0, stream>>>(logits, diff, out);
}